// PatchMix_78237124264127
// MI455X (gfx1250) — compile-verified
//
#include <hip/hip_runtime.h>
#include <hip/hip_bf16.h>
#include <cstddef>

// Problem geometry (compile-time constants from the reference).
#define B_  256
#define T_  196
#define C_  768
#define G_  32
#define M_  4
#define N_  (B_ / G_)                    // 8
#define S_  (T_ / M_)                    // 49
#define L_  (G_ * M_)                    // 128
#define CHUNK_FLOATS (S_ * C_)           // 37632 contiguous floats per (n,l) chunk
#define CHUNK_F4     (CHUNK_FLOATS / 4)  // 9408 x 16B units per chunk
#define OUT_MAIN     (B_ * T_ * C_)      // 38,535,168 floats
#define NTGT         (B_ * M_)           // 1024 target entries
#define DEPTH 8                          // async ops in flight per wave
#define TPB   256                        // threads per block (8 waves)

// CDNA5 async global<->LDS data-mover path (ASYNCcnt-tracked), with a
// plain-B128 fallback if the builtins are absent on this toolchain.
#if defined(__AMDGCN__) && defined(__gfx1250__) && \
    __has_builtin(__builtin_amdgcn_global_load_async_to_lds_b128) && \
    __has_builtin(__builtin_amdgcn_global_store_async_from_lds_b128) && \
    __has_builtin(__builtin_amdgcn_s_wait_asynccnt)
#define USE_ASYNC 1
#else
#define USE_ASYNC 0
#endif

#if USE_ASYNC
// Builtin signature (per hipcc diagnostic): addrspace(1) int4* global side,
// addrspace(3) int4* LDS side, then two immediate ints (offset, cpol).
typedef int v4i_ __attribute__((ext_vector_type(4)));
typedef __attribute__((address_space(1))) v4i_ g_v4i;   // global int4
typedef __attribute__((address_space(3))) v4i_ l_v4i;   // LDS int4
#define CPOL_NT 1  // TH=NT: single-pass stream, don't thrash the 192MB L2
#endif

// One block per chunk: copies 150,528 contiguous bytes from the permuted
// source chunk to the destination chunk via the async data mover, with a
// DEPTH-deep batched pipeline per wave (loads phase / stores phase, since
// ASYNCcnt decrements are unordered between loads and stores).
__global__ __launch_bounds__(TPB) void PatchMix_copy_kernel(
    const float* __restrict__ src, float* __restrict__ dst) {
  const int o  = blockIdx.x;            // 0..1023 = n*L + l_out
  const int n  = o >> 7;                // / L_
  const int lo = o & (L_ - 1);          // l_out
  // idx[l] = (l + (l % m) * m) % L
  const int ls = (lo + ((lo & (M_ - 1)) << 2)) & (L_ - 1);

  const size_t soff =
      ((size_t)(n * G_ + (ls >> 2)) * T_ + (size_t)((ls & 3) * S_)) * C_;
  const size_t doff =
      ((size_t)(n * G_ + (lo >> 2)) * T_ + (size_t)((lo & 3) * S_)) * C_;

  const float4* __restrict__ s4 = (const float4*)(src + soff);
  float4*       __restrict__ d4 = (float4*)(dst + doff);
  const int tid = threadIdx.x;

#if USE_ASYNC
  __shared__ float4 stage[TPB * DEPTH];   // 32 KB: DEPTH 16B slots per thread
  // 5 outer iterations: base = 0,2048,...,8192 (CHUNK_F4 = 9408).
  for (int base = 0; base < CHUNK_F4; base += TPB * DEPTH) {
    // Phase 1: issue up to DEPTH async loads (ASYNCcnt ramps to DEPTH).
#pragma unroll
    for (int k = 0; k < DEPTH; ++k) {
      const int i = base + k * TPB + tid;
      if (i < CHUNK_F4) {
        __builtin_amdgcn_global_load_async_to_lds_b128(
            (g_v4i*)(s4 + i), (l_v4i*)&stage[k * TPB + tid], 0, CPOL_NT);
      }
    }
    __builtin_amdgcn_s_wait_asynccnt(0);  // all loads landed in LDS
    // Phase 2: issue up to DEPTH async stores from the same slots.
#pragma unroll
    for (int k = 0; k < DEPTH; ++k) {
      const int i = base + k * TPB + tid;
      if (i < CHUNK_F4) {
        __builtin_amdgcn_global_store_async_from_lds_b128(
            (g_v4i*)(d4 + i), (l_v4i*)&stage[k * TPB + tid], 0, CPOL_NT);
      }
    }
    __builtin_amdgcn_s_wait_asynccnt(0);  // stores consumed LDS slots
  }
#else
  // Fallback: direct B128 vector copy (global_load_b128/global_store_b128).
  for (int i = tid; i < CHUNK_F4; i += TPB) {
    d4[i] = s4[i];
  }
#endif
}

// target[b, k] = (b // g) * g + ((b % g + k) % g), written as float
// (d_out is float* for this tuple; values are small integers, exact in f32).
__global__ __launch_bounds__(TPB) void PatchMix_target_kernel(
    float* __restrict__ tgt) {
  const int j = blockIdx.x * blockDim.x + threadIdx.x;
  if (j < NTGT) {
    const int b = j >> 2;
    const int k = j & 3;
    const int v = (b & ~(G_ - 1)) + (((b & (G_ - 1)) + k) & (G_ - 1));
    tgt[j] = (float)v;
  }
}

extern "C" void kernel_launch(void* const* d_in, const int* in_sizes, int n_in,
                              void* d_out, int out_size, void* d_ws, size_t ws_size,
                              hipStream_t stream) {
  (void)in_sizes; (void)n_in; (void)out_size; (void)d_ws; (void)ws_size;
  const float* patches = (const float*)d_in[0];
  float* out = (float*)d_out;

  // 1024 chunks of 150,528 contiguous bytes each.
  PatchMix_copy_kernel<<<N_ * L_, TPB, 0, stream>>>(patches, out);
  // 1024 target entries appended after the main output.
  PatchMix_target_kernel<<<(NTGT + 255) / 256, 256, 0, stream>>>(out + OUT_MAIN);
}